// rain_degradation_88768384074353
// MI455X (gfx1250) — compile-verified
//
#include <hip/hip_runtime.h>
#include <hip/hip_bf16.h>

// Problem constants (fixed by setup_inputs)
constexpr int B  = 4;
constexpr int C  = 3;
constexpr int H  = 512;
constexpr int W  = 512;
constexpr int N  = 8;     // branches (original + 7 degraded)
constexpr int LH = 32;
constexpr int LW = 32;
constexpr int HW = H * W; // 262144 = 1<<18
constexpr float INV_SCALE = 1.0f / 16.0f;  // LH/H

__global__ __launch_bounds__(256)
void rain_degradation_88768384074353_kernel(
    const float* __restrict__ original,   // [B,C,H,W]
    const float* __restrict__ degraded,   // [B,N-1,C,H,W]
    const float* __restrict__ logits,     // [N,LH,LW]
    const int*   __restrict__ topk_ptr,   // scalar
    float*       __restrict__ out)        // [B,C,H,W]
{
    // ---- stage all low-res logits (8*32*32*4 = 32 KB) into LDS via
    //      CDNA5 async global->LDS DMA (ASYNCcnt-tracked) ----
    __shared__ __align__(16) float slog[N * LH * LW];

    const int tid = threadIdx.x;
    #pragma unroll
    for (int i = 0; i < 8; ++i) {
        const int v4 = i * 256 + tid;  // float4 index, 0..2047
        unsigned lds_off =
            (unsigned)(unsigned long long)(&slog[v4 * 4]);
        unsigned long long gaddr =
            (unsigned long long)(const void*)(logits + v4 * 4);
        asm volatile("global_load_async_to_lds_b128 %0, %1, off"
                     :: "v"(lds_off), "v"(gaddr) : "memory");
    }
    asm volatile("s_wait_asynccnt 0x0" ::: "memory");
    __syncthreads();

    // ---- this thread's pixel ----
    const int p = blockIdx.x * 256 + tid;   // 0 .. HW-1 (grid = HW/256)
    const int h = p >> 9;                   // / W
    const int w = p & (W - 1);

    // Bilinear upsample coords: half-pixel centers, edge clamp.
    // (For a triangle kernel, jax.image.resize's drop-OOB-tap + renormalize
    //  is numerically identical to index clamping.)
    const float sy = ((float)h + 0.5f) * INV_SCALE - 0.5f;
    const float sx = ((float)w + 0.5f) * INV_SCALE - 0.5f;
    const float fy0 = floorf(sy), fx0 = floorf(sx);
    const float ty = sy - fy0,    tx = sx - fx0;
    int y0 = (int)fy0, x0 = (int)fx0;
    int y1 = y0 + 1,   x1 = x0 + 1;
    y0 = max(0, min(LH - 1, y0));  y1 = max(0, min(LH - 1, y1));
    x0 = max(0, min(LW - 1, x0));  x1 = max(0, min(LW - 1, x1));
    const float w00 = (1.0f - ty) * (1.0f - tx);
    const float w01 = (1.0f - ty) * tx;
    const float w10 = ty * (1.0f - tx);
    const float w11 = ty * tx;

    float up[N];
    #pragma unroll
    for (int n = 0; n < N; ++n) {
        const float* Ln = &slog[n * (LH * LW)];
        up[n] = w00 * Ln[y0 * LW + x0] + w01 * Ln[y0 * LW + x1]
              + w10 * Ln[y1 * LW + x0] + w11 * Ln[y1 * LW + x1];
    }

    // ---- softmax over branches (TEMPERATURE = 1) ----
    float m = up[0];
    #pragma unroll
    for (int n = 1; n < N; ++n) m = fmaxf(m, up[n]);
    float wv[N];
    float s = 0.0f;
    #pragma unroll
    for (int n = 0; n < N; ++n) { wv[n] = __expf(up[n] - m); s += wv[n]; }
    const float invs = 1.0f / s;
    #pragma unroll
    for (int n = 0; n < N; ++n) wv[n] *= invs;

    // ---- top-k gate: k-th largest as threshold, keep >= thr (tie-exact) ----
    int kk = topk_ptr[0];
    kk = kk < 1 ? 1 : (kk > N ? N : kk);
    float thr = 0.0f;
    {
        float tmp[N];
        #pragma unroll
        for (int n = 0; n < N; ++n) tmp[n] = wv[n];
        for (int j = 0; j < kk; ++j) {
            float mx = -1.0f;
            #pragma unroll
            for (int n = 0; n < N; ++n) mx = fmaxf(mx, tmp[n]);
            thr = mx;
            bool removed = false;   // remove exactly one instance of mx
            #pragma unroll
            for (int n = 0; n < N; ++n) {
                if (!removed && tmp[n] == mx) { tmp[n] = -1.0f; removed = true; }
            }
        }
    }
    float s2 = 0.0f;
    #pragma unroll
    for (int n = 0; n < N; ++n) {
        if (wv[n] >= thr) s2 += wv[n]; else wv[n] = 0.0f;
    }
    const float inv2 = 1.0f / s2;
    #pragma unroll
    for (int n = 0; n < N; ++n) wv[n] *= inv2;

    // ---- weighted combine: only selected branches are loaded.
    //      Branch predicate hoisted outside the B*C loop -> 8 exec-maskable
    //      regions of 12 coalesced loads; unselected branches fetch nothing.
    float acc[B * C];
    #pragma unroll
    for (int i = 0; i < B * C; ++i) acc[i] = 0.0f;

    const float* orig_p = original + p;
    const float* degr_p = degraded + p;
    #pragma unroll
    for (int n = 0; n < N; ++n) {
        if (wv[n] != 0.0f) {
            const float wn = wv[n];
            #pragma unroll
            for (int b = 0; b < B; ++b) {
                #pragma unroll
                for (int c = 0; c < C; ++c) {
                    const float* src = (n == 0)
                        ? (orig_p + (size_t)(b * C + c) * HW)
                        : (degr_p + (size_t)((b * (N - 1) + (n - 1)) * C + c) * HW);
                    acc[b * C + c] = fmaf(wn, *src, acc[b * C + c]);
                }
            }
        }
    }

    #pragma unroll
    for (int b = 0; b < B; ++b)
        #pragma unroll
        for (int c = 0; c < C; ++c)
            out[(size_t)(b * C + c) * HW + p] = acc[b * C + c];
}

extern "C" void kernel_launch(void* const* d_in, const int* in_sizes, int n_in,
                              void* d_out, int out_size, void* d_ws, size_t ws_size,
                              hipStream_t stream) {
    (void)in_sizes; (void)n_in; (void)out_size; (void)d_ws; (void)ws_size;
    const float* original = (const float*)d_in[0];
    const float* degraded = (const float*)d_in[1];
    const float* logits   = (const float*)d_in[2];
    const int*   topk     = (const int*)d_in[3];
    float* out = (float*)d_out;

    dim3 grid(HW / 256);   // 1024 blocks
    dim3 block(256);       // 8 wave32s
    rain_degradation_88768384074353_kernel<<<grid, block, 0, stream>>>(
        original, degraded, logits, topk, out);
}